// TypeNorm_9887014716159
// MI455X (gfx1250) — compile-verified
//
#include <hip/hip_runtime.h>
#include <stdint.h>

typedef __attribute__((ext_vector_type(2))) float v2f;
typedef __attribute__((ext_vector_type(8))) float v8f;

#define DIM     240
#define NSCAL   64
#define NFEAT   112
#define NTYPE   10
#define NCOLS   192      // padded stat columns: [0,64)=S1 [64,128)=S2 [128,160)=SV [160,176)=SW [176]=count
#define CNT_COL 176
#define EPS     1e-5f

#define GSTATS_F 3072            // 16*192 floats reserved (10 used)
#define NPARAM   7200            // cmu[2400] | cscale[2400] | cbias[2400]

// ---------------- Pass 1: per-type stats via one-hot WMMA segment-sum ----------------
__global__ __launch_bounds__(256) void k_stats(const float* __restrict__ x,
                                               const int* __restrict__ types,
                                               float* __restrict__ gstats,
                                               int batch, int ntiles) {
    __shared__ float pt[32 * NCOLS];   // per-row stat contributions (B operand source)
    __shared__ int   tt[32];           // per-row types (A operand source)

    const int tid  = threadIdx.x;
    const int lane = tid & 31;
    const int wave = tid >> 5;
    const int m    = lane & 15;        // M (type) / N (column-in-tile) sub-index
    const int half = lane >> 4;        // 0: K rows {0,1}; 1: K rows {2,3}

    // 12 column tiles of 16. Wave w always computes tile w and tile 8+(w&3);
    // the second tile is redundantly computed by wave pairs {w, w+4} so both
    // WMMAs are unconditional (no EXEC masking around v_wmma); only waves 0-3
    // flush the second accumulator.
    const int n0a = wave * 16;
    const int n0b = (8 + (wave & 3)) * 16;
    v8f acc0 = {}; v8f acc1 = {};

    for (int tile = blockIdx.x; tile < ntiles; tile += gridDim.x) {
        const int rbase = tile * 32;

        // zero stat tile, load types (invalid rows -> padded type 15, discarded later)
        for (int i = tid; i < 32 * NCOLS; i += 256) pt[i] = 0.f;
        if (tid < 32) {
            int r = rbase + tid;
            tt[tid] = (r < batch) ? types[r] : 15;
        }
        __syncthreads();

        // fill per-row stat contributions; coalesced streaming read of x
        for (int i = tid; i < 32 * DIM; i += 256) {
            int r = i / DIM, c = i - r * DIM;
            int row = rbase + r;
            if (row < batch) {
                float v = x[(size_t)row * DIM + c];
                float* p = &pt[r * NCOLS];
                if (c < NSCAL) {                       // scalars: sum x, sum x^2
                    p[c]         = v;
                    p[NSCAL + c] = v * v;
                } else if (c < 160) {                  // 32x1o: mean of 3 comps of v^2
                    atomicAdd(&p[128 + (c - NSCAL) / 3], v * v * (1.f / 3.f));
                } else {                               // 16x2e: mean of 5 comps of v^2
                    atomicAdd(&p[160 + (c - 160) / 5], v * v * 0.2f);
                }
                if (c == 0) p[CNT_COL] = 1.f;          // count column
            }
        }
        __syncthreads();

        // D(16x16) += A(16x4 one-hot types) x B(4x16 stat cols), K=4 rows per step
        for (int r0 = 0; r0 < 32; r0 += 4) {
            int rA = r0 + 2 * half;    // K = 0 / 2
            int rB = rA + 1;           // K = 1 / 3
            v2f A;
            A.x = (tt[rA] == m) ? 1.f : 0.f;
            A.y = (tt[rB] == m) ? 1.f : 0.f;
            v2f B0, B1;
            B0.x = pt[rA * NCOLS + n0a + m];
            B0.y = pt[rB * NCOLS + n0a + m];
            B1.x = pt[rA * NCOLS + n0b + m];
            B1.y = pt[rB * NCOLS + n0b + m];
            acc0 = __builtin_amdgcn_wmma_f32_16x16x4_f32(false, A, false, B0,
                                                         (short)0, acc0, false, false);
            acc1 = __builtin_amdgcn_wmma_f32_16x16x4_f32(false, A, false, B1,
                                                         (short)0, acc1, false, false);
        }
        __syncthreads();
    }

    // flush wave-private accumulators: lane L, vgpr v -> (M = v + 8*(L>=16), N = n0 + L%16)
    #pragma unroll
    for (int v = 0; v < 8; ++v) {
        int mm = (lane < 16) ? v : (v + 8);
        int n  = n0a + m;
        if (mm < NTYPE && n <= CNT_COL) atomicAdd(&gstats[mm * NCOLS + n], acc0[v]);
    }
    if (wave < 4) {   // second tile flushed once (waves 4-7 computed a discarded duplicate)
        #pragma unroll
        for (int v = 0; v < 8; ++v) {
            int mm = (lane < 16) ? v : (v + 8);
            int n  = n0b + m;
            if (mm < NTYPE && n <= CNT_COL) atomicAdd(&gstats[mm * NCOLS + n], acc1[v]);
        }
    }
}

// ---------------- Pass 2: finalize -> expanded per-column mu / scale / bias ----------------
__global__ void k_finalize(const float* __restrict__ gstats,
                           const float* __restrict__ weight,
                           const float* __restrict__ bias,
                           float* __restrict__ cmu,
                           float* __restrict__ csc,
                           float* __restrict__ cbi) {
    int idx = blockIdx.x * blockDim.x + threadIdx.x;
    if (idx >= NTYPE * DIM) return;
    int t = idx / DIM, c = idx - t * DIM;
    const float* gs = &gstats[t * NCOLS];
    float inv = 1.f / fmaxf(gs[CNT_COL], 1.f);
    float mu = 0.f, sc, bi = 0.f;
    if (c < NSCAL) {
        mu = gs[c] * inv;
        float var = gs[NSCAL + c] * inv - mu * mu;   // E[(x-mu)^2] = E[x^2] - mu^2
        sc = rsqrtf(var + EPS) * weight[t * NFEAT + c];
        bi = bias[t * NSCAL + c];
    } else if (c < 160) {
        int f = (c - NSCAL) / 3;
        sc = rsqrtf(gs[128 + f] * inv + EPS) * weight[t * NFEAT + NSCAL + f];
    } else {
        int f = (c - 160) / 5;
        sc = rsqrtf(gs[160 + f] * inv + EPS) * weight[t * NFEAT + 96 + f];
    }
    cmu[idx] = mu; csc[idx] = sc; cbi[idx] = bi;
}

// ---------------- Pass 3: apply (streaming float4), params staged via async-to-LDS ----------------
__global__ __launch_bounds__(256) void k_apply(const float* __restrict__ x,
                                               const int* __restrict__ types,
                                               const float* __restrict__ params, // cmu|csc|cbi
                                               float* __restrict__ out,
                                               int batch) {
    __shared__ __align__(16) float sp[NPARAM];

    // stage 28.8KB of per-column params with gfx1250 async copies
    for (int i = threadIdx.x * 4; i < NPARAM; i += 256 * 4) {
        uint32_t lds_off = (uint32_t)(uintptr_t)(&sp[i]);  // low 32 bits of flat shared addr = LDS offset
        const float* g = params + i;
        asm volatile("global_load_async_to_lds_b128 %0, %1, off"
                     :: "v"(lds_off), "v"(g) : "memory");
    }
    asm volatile("s_wait_asynccnt 0" ::: "memory");
    __syncthreads();

    const int nvec   = batch * (DIM / 4);
    const int stride = gridDim.x * blockDim.x;
    for (int i = blockIdx.x * blockDim.x + threadIdx.x; i < nvec; i += stride) {
        int row = i / (DIM / 4);
        int c0  = (i - row * (DIM / 4)) * 4;
        int t   = types[row];
        int pb  = t * DIM + c0;
        float4 xv = ((const float4*)x)[i];
        float4 mu = *(const float4*)&sp[pb];
        float4 sc = *(const float4*)&sp[2400 + pb];
        float4 bi = *(const float4*)&sp[4800 + pb];
        float4 o;
        o.x = (xv.x - mu.x) * sc.x + bi.x;
        o.y = (xv.y - mu.y) * sc.y + bi.y;
        o.z = (xv.z - mu.z) * sc.z + bi.z;
        o.w = (xv.w - mu.w) * sc.w + bi.w;
        ((float4*)out)[i] = o;
    }
}

extern "C" void kernel_launch(void* const* d_in, const int* in_sizes, int n_in,
                              void* d_out, int out_size, void* d_ws, size_t ws_size,
                              hipStream_t stream) {
    const float* x  = (const float*)d_in[0];
    const int*   ty = (const int*)d_in[1];
    const float* w  = (const float*)d_in[2];
    const float* b  = (const float*)d_in[3];
    float* out = (float*)d_out;
    const int batch = in_sizes[1];

    float* ws     = (float*)d_ws;
    float* gstats = ws;                    // 3072 floats, atomically accumulated
    float* cmu    = ws + GSTATS_F;         // 2400
    float* csc    = cmu + 2400;            // 2400
    float* cbi    = csc + 2400;            // 2400

    hipMemsetAsync(gstats, 0, GSTATS_F * sizeof(float), stream);

    int ntiles = (batch + 31) / 32;
    int g1 = ntiles < 1024 ? ntiles : 1024;
    k_stats<<<g1, 256, 0, stream>>>(x, ty, gstats, batch, ntiles);

    k_finalize<<<(NTYPE * DIM + 255) / 256, 256, 0, stream>>>(gstats, w, b, cmu, csc, cbi);

    k_apply<<<2048, 256, 0, stream>>>(x, ty, cmu, out, batch);
}